// MultiHeadAttention_7361573945686
// MI455X (gfx1250) — compile-verified
//
#include <hip/hip_runtime.h>
#include <math.h>

#define NN 50000
#define NE 800000
#define DN 128
#define DE 64
#define DKV 192
#define NH 8
#define HD 16

typedef __attribute__((ext_vector_type(2))) float v2f;
typedef __attribute__((ext_vector_type(8))) float v8f;

__device__ __forceinline__ unsigned ord_enc(float x) {
    unsigned u = __float_as_uint(x);
    return (u & 0x80000000u) ? ~u : (u | 0x80000000u);
}
__device__ __forceinline__ float ord_dec(unsigned u) {
    return (u & 0x80000000u) ? __uint_as_float(u ^ 0x80000000u) : __uint_as_float(~u);
}

// Async global->LDS copy of 16 bytes (GV mode), tracked by ASYNCcnt.
__device__ __forceinline__ void async_g2l_b128(unsigned lds_byte_off, const void* gaddr) {
    asm volatile("global_load_async_to_lds_b128 %0, %1, off"
                 :: "v"(lds_byte_off), "v"((unsigned long long)(size_t)gaddr)
                 : "memory");
}
__device__ __forceinline__ void wait_async0() {
    asm volatile("s_wait_asynccnt 0x0" ::: "memory");
}

// ---------------- init accumulators ----------------
__global__ void k_init(unsigned* __restrict__ segmax, float* __restrict__ segsum,
                       float* __restrict__ obuf) {
    long i = (long)blockIdx.x * blockDim.x + threadIdx.x;
    if (i < (long)NN * DN) obuf[i] = 0.0f;
    if (i < (long)NN * NH) { segsum[i] = 0.0f; segmax[i] = 0u; }
}

// ---------------- fused QKV + logits ----------------
#define TILE_E 64
#define STR 196   // LDS row stride (floats); 196 % 64 == 4 -> conflict-free A frags

__global__ __launch_bounds__(256) void k_qkv(
    const float* __restrict__ node_feats, const float* __restrict__ edge_feats,
    const int* __restrict__ src, const int* __restrict__ tgt,
    const float* __restrict__ Wq, const float* __restrict__ bq,
    const float* __restrict__ Wk, const float* __restrict__ bk,
    const float* __restrict__ Wv, const float* __restrict__ bv,
    float* __restrict__ a_buf, float* __restrict__ v_buf,
    unsigned* __restrict__ segmax)
{
    __shared__ float lds[TILE_E * STR];

    const int tid  = threadIdx.x;
    const int lane = tid & 31;
    const int h    = tid >> 5;      // wave id == head id
    const int kg   = lane >> 4;     // k-group half (0/1)
    const int lm   = lane & 15;     // M (A frag) / N (B,C frag)
    const long base = (long)blockIdx.x * TILE_E;
    const int col = h * HD + lm;    // output column this lane owns

    // ---- stage A_q = node_feats[tgt[e]]  (64 x 128) via async global->LDS ----
    #pragma unroll
    for (int it = 0; it < 8; ++it) {
        int idx = tid + it * 256;        // 64 rows * 32 float4
        int row = idx >> 5;
        int c4  = idx & 31;
        long e  = base + row;
        const float* g = node_feats + (long)tgt[e] * DN + c4 * 4;
        async_g2l_b128((unsigned)(size_t)&lds[row * STR + c4 * 4], g);
    }
    wait_async0();
    __syncthreads();

    // ---- Q = A_q @ Wq[:, head cols] + bq ----
    v8f cq[4];
    {
        float b = bq[col];
        #pragma unroll
        for (int ms = 0; ms < 4; ++ms)
            #pragma unroll
            for (int r = 0; r < 8; ++r) cq[ms][r] = b;
    }
    #pragma unroll 4
    for (int t = 0; t < DN / 4; ++t) {
        int kk = 4 * t;
        v2f b;
        b.x = Wq[(kk + 2 * kg + 0) * DN + col];
        b.y = Wq[(kk + 2 * kg + 1) * DN + col];
        #pragma unroll
        for (int ms = 0; ms < 4; ++ms) {
            v2f a = *(const v2f*)&lds[(ms * 16 + lm) * STR + kk + 2 * kg];
            cq[ms] = __builtin_amdgcn_wmma_f32_16x16x4_f32(
                false, a, false, b, (short)0, cq[ms], false, false);
        }
    }
    __syncthreads();

    // ---- stage A_kv = [node_feats[src[e]] || edge_feats[e]]  (64 x 192) async ----
    #pragma unroll
    for (int it = 0; it < 12; ++it) {
        int idx = tid + it * 256;        // 64 rows * 48 float4
        int row = idx / 48;
        int c4  = idx % 48;
        long e  = base + row;
        const float* g;
        if (c4 < 32) g = node_feats + (long)src[e] * DN + c4 * 4;
        else         g = edge_feats + e * DE + (c4 - 32) * 4;
        async_g2l_b128((unsigned)(size_t)&lds[row * STR + c4 * 4], g);
    }
    wait_async0();
    __syncthreads();

    // ---- K,V = A_kv @ {Wk,Wv}[:, head cols] + {bk,bv} ----
    v8f ck[4], cv[4];
    {
        float b0 = bk[col], b1 = bv[col];
        #pragma unroll
        for (int ms = 0; ms < 4; ++ms)
            #pragma unroll
            for (int r = 0; r < 8; ++r) { ck[ms][r] = b0; cv[ms][r] = b1; }
    }
    #pragma unroll 4
    for (int t = 0; t < DKV / 4; ++t) {
        int kk = 4 * t;
        v2f bkf, bvf;
        bkf.x = Wk[(kk + 2 * kg + 0) * DN + col];
        bkf.y = Wk[(kk + 2 * kg + 1) * DN + col];
        bvf.x = Wv[(kk + 2 * kg + 0) * DN + col];
        bvf.y = Wv[(kk + 2 * kg + 1) * DN + col];
        #pragma unroll
        for (int ms = 0; ms < 4; ++ms) {
            v2f a = *(const v2f*)&lds[(ms * 16 + lm) * STR + kk + 2 * kg];
            ck[ms] = __builtin_amdgcn_wmma_f32_16x16x4_f32(
                false, a, false, bkf, (short)0, ck[ms], false, false);
            cv[ms] = __builtin_amdgcn_wmma_f32_16x16x4_f32(
                false, a, false, bvf, (short)0, cv[ms], false, false);
        }
    }

    // ---- logits a[e,h] = (q . k) / sqrt(HD); seg-max; store V ----
    const float scale = 0.25f;
    #pragma unroll
    for (int ms = 0; ms < 4; ++ms) {
        #pragma unroll
        for (int r = 0; r < 8; ++r) {
            float s = cq[ms][r] * ck[ms][r];
            // reduce over the 16 head-dim lanes within each half
            #pragma unroll
            for (int off = 1; off < 16; off <<= 1) s += __shfl_xor(s, off, 32);
            s *= scale;
            if (lm == r) {
                long e = base + ms * 16 + r + 8 * kg;
                a_buf[e * NH + h] = s;
                atomicMax(&segmax[(long)tgt[e] * NH + h], ord_enc(s));
            }
        }
        #pragma unroll
        for (int r = 0; r < 8; ++r) {
            long e = base + ms * 16 + r + 8 * kg;
            v_buf[e * DN + h * HD + lm] = cv[ms][r];
        }
    }
}

// ---------------- exp + segment sum ----------------
__global__ void k_expsum(const int* __restrict__ tgt, float* __restrict__ a_buf,
                         const unsigned* __restrict__ segmax, float* __restrict__ segsum) {
    long i = (long)blockIdx.x * blockDim.x + threadIdx.x;
    if (i >= (long)NE * NH) return;
    long e = i >> 3;
    int  h = (int)(i & 7);
    long t = tgt[e];
    float m = ord_dec(segmax[t * NH + h]);
    float w = __expf(a_buf[i] - m);
    a_buf[i] = w;
    atomicAdd(&segsum[t * NH + h], w);
}

// ---------------- normalized weighted scatter aggregate ----------------
__global__ void k_aggregate(const int* __restrict__ tgt, const float* __restrict__ a_buf,
                            const float* __restrict__ segsum, const float* __restrict__ v_buf,
                            float* __restrict__ obuf) {
    long i = (long)blockIdx.x * blockDim.x + threadIdx.x;
    if (i >= (long)NE * DN) return;
    long e = i >> 7;
    int  c = (int)(i & 127);
    int  h = c >> 4;
    long t = tgt[e];
    float w = a_buf[e * NH + h] / segsum[t * NH + h];
    atomicAdd(&obuf[t * DN + c], w * v_buf[i]);
}

// ---------------- output GEMM: out = o @ Wo + bo ----------------
#define OSTR 132  // 132 % 64 == 4 -> conflict-free

__global__ __launch_bounds__(256) void k_out(
    const float* __restrict__ obuf, const float* __restrict__ Wo,
    const float* __restrict__ bo, float* __restrict__ out)
{
    __shared__ float lds[TILE_E * OSTR];
    const int tid  = threadIdx.x;
    const int lane = tid & 31;
    const int h    = tid >> 5;
    const int kg   = lane >> 4;
    const int lm   = lane & 15;
    const long base = (long)blockIdx.x * TILE_E;
    const int col = h * HD + lm;

    #pragma unroll
    for (int it = 0; it < 8; ++it) {
        int idx = tid + it * 256;
        int row = idx >> 5;
        int c4  = idx & 31;
        long r  = base + row;
        if (r < NN) {
            async_g2l_b128((unsigned)(size_t)&lds[row * OSTR + c4 * 4],
                           obuf + r * DN + c4 * 4);
        } else {
            *(float4*)&lds[row * OSTR + c4 * 4] = make_float4(0.f, 0.f, 0.f, 0.f);
        }
    }
    wait_async0();
    __syncthreads();

    v8f c[4];
    {
        float b = bo[col];
        #pragma unroll
        for (int ms = 0; ms < 4; ++ms)
            #pragma unroll
            for (int r = 0; r < 8; ++r) c[ms][r] = b;
    }
    #pragma unroll 4
    for (int t = 0; t < DN / 4; ++t) {
        int kk = 4 * t;
        v2f b;
        b.x = Wo[(kk + 2 * kg + 0) * DN + col];
        b.y = Wo[(kk + 2 * kg + 1) * DN + col];
        #pragma unroll
        for (int ms = 0; ms < 4; ++ms) {
            v2f a = *(const v2f*)&lds[(ms * 16 + lm) * OSTR + kk + 2 * kg];
            c[ms] = __builtin_amdgcn_wmma_f32_16x16x4_f32(
                false, a, false, b, (short)0, c[ms], false, false);
        }
    }
    #pragma unroll
    for (int ms = 0; ms < 4; ++ms)
        #pragma unroll
        for (int r = 0; r < 8; ++r) {
            long row = base + ms * 16 + r + 8 * kg;
            if (row < NN) out[row * DN + col] = c[ms][r];
        }
}

// ---------------- launch ----------------
extern "C" void kernel_launch(void* const* d_in, const int* in_sizes, int n_in,
                              void* d_out, int out_size, void* d_ws, size_t ws_size,
                              hipStream_t stream) {
    const float* node_feats = (const float*)d_in[0];
    const float* edge_feats = (const float*)d_in[1];
    const int*   edge_index = (const int*)d_in[2];
    const float* Wq = (const float*)d_in[3];
    const float* bq = (const float*)d_in[4];
    const float* Wk = (const float*)d_in[5];
    const float* bk = (const float*)d_in[6];
    const float* Wv = (const float*)d_in[7];
    const float* bv = (const float*)d_in[8];
    const float* Wo = (const float*)d_in[9];
    const float* bo = (const float*)d_in[10];
    float* out = (float*)d_out;

    const int* src = edge_index;
    const int* tgt = edge_index + NE;

    char* ws = (char*)d_ws;
    unsigned* segmax = (unsigned*)(ws + 0);                 // N*8 u32   = 1.6 MB
    float*    segsum = (float*)(ws + 1600000);              // N*8 f32   = 1.6 MB
    float*    a_buf  = (float*)(ws + 3200000);              // E*8 f32   = 25.6 MB
    float*    obuf   = (float*)(ws + 28800000);             // N*128 f32 = 25.6 MB
    float*    v_buf  = (float*)(ws + 54400000);             // E*128 f32 = 409.6 MB

    k_init<<<(NN * DN + 255) / 256, 256, 0, stream>>>(segmax, segsum, obuf);

    k_qkv<<<NE / TILE_E, 256, 0, stream>>>(
        node_feats, edge_feats, src, tgt,
        Wq, bq, Wk, bk, Wv, bv, a_buf, v_buf, segmax);

    k_expsum<<<((long)NE * NH + 255) / 256, 256, 0, stream>>>(tgt, a_buf, segmax, segsum);

    k_aggregate<<<((long)NE * DN + 255) / 256, 256, 0, stream>>>(tgt, a_buf, segsum, v_buf, obuf);

    k_out<<<(NN + TILE_E - 1) / TILE_E, 256, 0, stream>>>(obuf, Wo, bo, out);
}